// OrdFilter_26886495273257
// MI455X (gfx1250) — compile-verified
//
#include <hip/hip_runtime.h>

// Problem constants (match the JAX reference).
#define IMG_H   512
#define IMG_W   512
#define PAD     2          // SIZE // 2, SIZE = 5
#define WIN     5
#define KTOP    9

// Tiling: 32x32 outputs per workgroup, 36x36 input tile in LDS.
#define TILE    32
#define TW      (TILE + 2 * PAD)     // 36
#define NTOT    (TW * TW)            // 1296
#define NRING   (NTOT - TILE * TILE) // 272 halo elements
#define NTHREADS 256                 // 8 wave32s

typedef unsigned int u32x4 __attribute__((ext_vector_type(4)));
typedef int          i32x4 __attribute__((ext_vector_type(4)));
typedef int          i32x8 __attribute__((ext_vector_type(8)));

__device__ __forceinline__ int reflect512(int v) {
    // ReflectionPad2d semantics: -1 -> 1, -2 -> 2; 512 -> 510, 513 -> 509.
    v = (v < 0) ? -v : v;
    v = (v > IMG_H - 1) ? (2 * (IMG_H - 1) - v) : v;
    return v;
}

__global__ __launch_bounds__(NTHREADS) void OrdFilter_26886495273257_kernel(
        const float* __restrict__ inp, float* __restrict__ out) {
    __shared__ float tile[NTOT];

    const int tid = threadIdx.x;
    const int bx  = blockIdx.x * TILE;   // output tile origin x
    const int by  = blockIdx.y * TILE;   // output tile origin y
    const int b   = blockIdx.z;          // batch index

    const float* __restrict__ src = inp + (size_t)b * (IMG_H * IMG_W);

    // LDS byte offset of `tile` within the workgroup's LDS allocation:
    // generic pointers to LDS carry the group-segment offset in their low 32 bits.
    const unsigned lds_base = (unsigned)(size_t)(&tile[0]);

    // ============================================================
    // Stage 1a: TDM bulk DMA of the always-in-bounds 32x32 interior.
    // Rows land at LDS stride 36 floats via the TDM pad feature:
    //   pad_interval=4 -> pad after every 32 DWORDs (128B)
    //   pad_amount=3   -> 4 DWORDs (16B) of padding  => row stride 144B.
    // Issued once, by wave 0 only (EXEC is ignored by tensor ops).
    // ============================================================
    if ((tid >> 5) == 0) {
        const unsigned long long gaddr =
            (unsigned long long)(src + (size_t)by * IMG_W + bx);

        u32x4 g0;
        g0[0] = 1u;                                      // count=1 valid descriptor
        g0[1] = lds_base + (unsigned)((PAD * TW + PAD) * 4); // lds_addr (interior origin)
        g0[2] = (unsigned)(gaddr & 0xFFFFFFFFu);         // global_addr[31:0]
        g0[3] = (unsigned)((gaddr >> 32) & 0x01FFFFFFu)  // global_addr[56:32]
              | (2u << 30);                              // type=2 ("image")

        i32x8 g1;
        g1[0] = (2 << 16)      // data_size = 4 bytes
              | (1 << 20)      // pad_enable
              | (4 << 22)      // pad_interval: 2^(4+1)=32 DWORDs per row
              | (3 << 25);     // pad_amount: 3+1=4 DWORDs of padding
        g1[1] = (int)(512u << 16);  // tensor_dim0[15:0]=512 (no OOB for X<512)
        g1[2] = (int)(512u << 16);  // tensor_dim0 hi=0 | tensor_dim1[15:0]=512
        g1[3] = (int)((unsigned)TILE << 16); // tensor_dim1 hi=0 | tile_dim0=32
        g1[4] = TILE;               // tile_dim1=32, tile_dim2=0 (2D)
        g1[5] = IMG_W;              // tensor_dim0_stride[31:0] = 512 elements
        g1[6] = 0;                  // stride0 hi, stride1 lo (unused for 2D)
        g1[7] = 0;

        i32x4 z4 = {0, 0, 0, 0};
#if __has_include(<hip/amd_detail/amd_gfx1250_TDM.h>)
        i32x8 z8 = {0, 0, 0, 0, 0, 0, 0, 0};
        __builtin_amdgcn_tensor_load_to_lds(g0, g1, z4, z4, z8, 0);
#else
        __builtin_amdgcn_tensor_load_to_lds(g0, g1, z4, z4, 0);
#endif
        __builtin_amdgcn_s_wait_tensorcnt(0);
    }

    // ============================================================
    // Stage 1b: halo ring (272 elements) with reflection folded into
    // per-lane addresses, via CDNA5 async global->LDS loads.
    // ============================================================
    for (int k = tid; k < NRING; k += NTHREADS) {
        int ty, tx;
        if (k < 144) {                       // top 2 + bottom 2 full rows
            const int base = (k >= 72) ? 34 : 0;
            const int k2   = (k >= 72) ? (k - 72) : k;
            const int row  = (k2 >= TW) ? 1 : 0;
            ty = base + row;
            tx = k2 - row * TW;
        } else {                             // side columns, rows 2..33
            const int k3 = k - 144;
            ty = 2 + (k3 >> 2);
            const int c = k3 & 3;            // 0,1 -> left; 2,3 -> right
            tx = (c < 2) ? c : (c + 32);     // 0,1,34,35
        }
        const int gy = reflect512(by - PAD + ty);
        const int gx = reflect512(bx - PAD + tx);
        const float* gptr = src + (size_t)gy * IMG_W + gx;
        const unsigned ldsoff = lds_base + (unsigned)(ty * TW + tx) * 4u;
        asm volatile("global_load_async_to_lds_b32 %0, %1, off"
                     :
                     : "v"(ldsoff), "v"(gptr)
                     : "memory");
    }
    asm volatile("s_wait_asynccnt 0x0" ::: "memory");
    __syncthreads();   // publishes TDM interior + all waves' halo writes

    // ============================================================
    // Stage 2: each thread owns a 4-tall output column.
    // ============================================================
    const int lx  = tid & (TILE - 1);        // 0..31
    const int ly0 = (tid >> 5) << 2;         // 0,4,...,28

    // 8x5 register block covering the 4 stacked 5x5 windows of this thread.
    float r[8][WIN];
#pragma unroll
    for (int j = 0; j < 8; ++j) {
#pragma unroll
        for (int i = 0; i < WIN; ++i) {
            r[j][i] = tile[(ly0 + j) * TW + (lx + i)];
        }
    }

    float* __restrict__ outp = out + (size_t)b * (IMG_H * IMG_W);

#pragma unroll
    for (int p = 0; p < 4; ++p) {
        float a[KTOP];   // sorted ascending, a[0] = smallest of current top set

        // Grow phase: branch-free insertion of the first 9 window values
        // into a sorted prefix (element j costs 2j min/max ops).
#pragma unroll
        for (int j = 0; j < KTOP; ++j) {
            float t = r[p + j / WIN][j % WIN];
#pragma unroll
            for (int i = j - 1; i >= 0; --i) {
                const float hi = fmaxf(a[i], t);
                t = fminf(a[i], t);
                a[i + 1] = hi;
            }
            a[0] = t;
        }

        // Steady state: insert remaining 16, dropping the smallest each time.
#pragma unroll
        for (int j = KTOP; j < WIN * WIN; ++j) {
            float t = r[p + j / WIN][j % WIN];
#pragma unroll
            for (int i = KTOP - 1; i >= 0; --i) {
                const float hi = fmaxf(a[i], t);
                t = fminf(a[i], t);
                a[i] = hi;
            }
        }

        float s = 0.0f;
#pragma unroll
        for (int i = 0; i < KTOP; ++i) s += a[i];

        outp[(size_t)(by + ly0 + p) * IMG_W + (bx + lx)] = s * (1.0f / 9.0f);
    }
}

extern "C" void kernel_launch(void* const* d_in, const int* in_sizes, int n_in,
                              void* d_out, int out_size, void* d_ws, size_t ws_size,
                              hipStream_t stream) {
    (void)n_in; (void)d_ws; (void)ws_size; (void)out_size;
    const float* inp = (const float*)d_in[0];
    float* out = (float*)d_out;

    const int batches = in_sizes[0] / (IMG_H * IMG_W);   // 16
    dim3 grid(IMG_W / TILE, IMG_H / TILE, batches);       // (16,16,16)
    dim3 block(NTHREADS);                                 // 8 wave32s

    OrdFilter_26886495273257_kernel<<<grid, block, 0, stream>>>(inp, out);
}